// GATEncoder_46729244180551
// MI455X (gfx1250) — compile-verified
//
#include <hip/hip_runtime.h>
#include <hip/hip_bf16.h>
#include <math.h>

#define NN    50000
#define EE    400000
#define ETOT  (EE + NN)
#define GG    64
#define INC   256
#define HID   128
#define OUTC  256
#define HEADS 4
#define HC1   (HEADS * HID)   // 512
#define LRELU 0.2f
#define LNEPS 1e-5f

typedef __attribute__((ext_vector_type(2))) float v2f;
typedef __attribute__((ext_vector_type(8))) float v8f;

// ---------------------------------------------------------------------------
// fp32 WMMA GEMM: C[M,N] = A[M,K] @ B[K,N], row-major, all dims multiples of
// 16 (M,N) / 4 (K). One wave32 computes one 16x16 tile via V_WMMA_F32_16X16X4_F32.
// Block = 128 threads = 4 waves covering 4 consecutive N-tiles.
// ---------------------------------------------------------------------------
__global__ void wmma_gemm_f32(const float* __restrict__ A,
                              const float* __restrict__ B,
                              float* __restrict__ C,
                              int M, int Nn, int K) {
    const int tid  = threadIdx.x;
    const int wave = tid >> 5;
    const int lane = tid & 31;
    const int half = lane >> 4;        // 0: lanes 0-15, 1: lanes 16-31
    const int l16  = lane & 15;
    const int tm = blockIdx.x;                                   // M tile
    const int tn = blockIdx.y * (blockDim.x >> 5) + wave;        // N tile
    const int row = tm * 16 + l16;     // A row this lane feeds (M = lane%16)
    const int col = tn * 16 + l16;     // B/C column this lane holds

    const float* Arow = A + (size_t)row * K;
    v8f acc = {};
    for (int kk = 0; kk < K; kk += 4) {
        // A 16x4: lanes 0-15 -> K = kk,kk+1 ; lanes 16-31 -> K = kk+2,kk+3
        v2f a = *(const v2f*)(Arow + kk + 2 * half);
        // B 4x16 mirrored: lane half selects K pair, l16 selects column
        const float* Bp = B + (size_t)(kk + 2 * half) * Nn + col;
        v2f b;
        b[0] = Bp[0];
        b[1] = Bp[Nn];
        acc = __builtin_amdgcn_wmma_f32_16x16x4_f32(
            /*neg_a=*/false, a, /*neg_b=*/false, b,
            /*c_mod=*/(short)0, acc, /*reuse_a=*/false, /*reuse_b=*/false);
    }
    // C/D layout: VGPR r holds M = r + 8*half, N = l16
    float* Crow = C + (size_t)(tm * 16 + half * 8) * Nn + col;
    #pragma unroll
    for (int r = 0; r < 8; ++r)
        Crow[(size_t)r * Nn] = acc[r];
}

// ---------------------------------------------------------------------------
// Per-(node, head) attention scores: es = <h, a_src>, ed = <h, a_dst>
// ---------------------------------------------------------------------------
__global__ void attn_scores(const float* __restrict__ h,
                            const float* __restrict__ a_src,
                            const float* __restrict__ a_dst,
                            float* __restrict__ es, float* __restrict__ ed,
                            int H, int C) {
    int t = blockIdx.x * blockDim.x + threadIdx.x;
    if (t >= NN * H) return;
    int n = t / H, hh = t - n * H;
    const float* hp = h + (size_t)n * H * C + (size_t)hh * C;
    const float* as = a_src + hh * C;
    const float* ad = a_dst + hh * C;
    float s0 = 0.f, s1 = 0.f;
    for (int c = 0; c < C; c += 4) {
        float4 hv = *(const float4*)(hp + c);
        float4 av = *(const float4*)(as + c);
        float4 dv = *(const float4*)(ad + c);
        s0 += hv.x * av.x + hv.y * av.y + hv.z * av.z + hv.w * av.w;
        s1 += hv.x * dv.x + hv.y * dv.y + hv.z * dv.z + hv.w * dv.w;
    }
    es[t] = s0;
    ed[t] = s1;
}

__global__ void init_ms(float* __restrict__ m, float* __restrict__ s, int n) {
    int t = blockIdx.x * blockDim.x + threadIdx.x;
    if (t < n) { m[t] = -__builtin_inff(); s[t] = 0.f; }
}

__device__ __forceinline__ int edge_src(const int* ei, int i) {
    return i < EE ? ei[i] : i - EE;
}
__device__ __forceinline__ int edge_dst(const int* ei, int i) {
    return i < EE ? ei[EE + i] : i - EE;
}

// float atomic max via monotone int/uint trick (works with -inf init)
__device__ __forceinline__ void atomicMaxF(float* addr, float v) {
    if (v >= 0.f) atomicMax((int*)addr, __float_as_int(v));
    else          atomicMin((unsigned int*)addr, __float_as_uint(v));
}

// Pass 1: e = leaky_relu(es[src]+ed[dst]); segment max over dst
__global__ void edge_logits(const int* __restrict__ ei,
                            const float* __restrict__ es,
                            const float* __restrict__ ed,
                            float* __restrict__ e, float* __restrict__ m, int H) {
    int i = blockIdx.x * blockDim.x + threadIdx.x;
    if (i >= ETOT) return;
    int s = edge_src(ei, i), d = edge_dst(ei, i);
    for (int hh = 0; hh < H; ++hh) {
        float v = es[s * H + hh] + ed[d * H + hh];
        v = v > 0.f ? v : LRELU * v;
        e[(size_t)i * H + hh] = v;
        atomicMaxF(&m[d * H + hh], v);
    }
}

// Pass 2: e <- exp(e - m[dst]); segment sum over dst
__global__ void edge_exp(const int* __restrict__ ei,
                         float* __restrict__ e,
                         const float* __restrict__ m,
                         float* __restrict__ ssum, int H) {
    int i = blockIdx.x * blockDim.x + threadIdx.x;
    if (i >= ETOT) return;
    int d = edge_dst(ei, i);
    for (int hh = 0; hh < H; ++hh) {
        float ex = expf(e[(size_t)i * H + hh] - m[d * H + hh]);
        e[(size_t)i * H + hh] = ex;
        atomicAdd(&ssum[d * H + hh], ex);
    }
}

// Pass 3: out[dst] += h[src] * alpha  (one wave32 per edge; 8 edges / block)
__global__ void edge_aggr(const int* __restrict__ ei,
                          const float* __restrict__ e,
                          const float* __restrict__ ssum,
                          const float* __restrict__ h,
                          float* __restrict__ out,
                          int H, int cshift /* log2(C) */) {
    int lane = threadIdx.x & 31;
    int i = blockIdx.x * (blockDim.x >> 5) + (threadIdx.x >> 5);
    if (i >= ETOT) return;
    int s = edge_src(ei, i), d = edge_dst(ei, i);
    float alpha[HEADS];
    for (int hh = 0; hh < H; ++hh)
        alpha[hh] = e[(size_t)i * H + hh] / (ssum[d * H + hh] + 1e-16f);
    int HCn = H << cshift;
    const float* hs = h + (size_t)s * HCn;
    float* od = out + (size_t)d * HCn;
    for (int c = lane; c < HCn; c += 32)
        atomicAdd(&od[c], hs[c] * alpha[c >> cshift]);
}

// (x + bias) -> LayerNorm -> optional ReLU, in place. One wave32 per row.
__global__ void ln_kernel(float* __restrict__ buf,
                          const float* __restrict__ bias,
                          const float* __restrict__ g,
                          const float* __restrict__ b,
                          int W, int relu) {
    int wave = threadIdx.x >> 5, lane = threadIdx.x & 31;
    int row = blockIdx.x * (blockDim.x >> 5) + wave;
    if (row >= NN) return;
    float* p = buf + (size_t)row * W;
    float vals[16];                      // W <= 512 -> up to 16 per lane
    int nv = W >> 5;
    float sum = 0.f, sq = 0.f;
    for (int j = 0; j < nv; ++j) {
        int c = lane + (j << 5);
        float v = p[c] + bias[c];
        vals[j] = v; sum += v; sq += v * v;
    }
    #pragma unroll
    for (int off = 16; off > 0; off >>= 1) {
        sum += __shfl_xor(sum, off, 32);
        sq  += __shfl_xor(sq,  off, 32);
    }
    float mu   = sum / (float)W;
    float var  = sq / (float)W - mu * mu;
    float rstd = rsqrtf(var + LNEPS);
    for (int j = 0; j < nv; ++j) {
        int c = lane + (j << 5);
        float v = (vals[j] - mu) * rstd * g[c] + b[c];
        if (relu) v = fmaxf(v, 0.f);
        p[c] = v;
    }
}

__global__ void pool_cnt(const int* __restrict__ batch, float* __restrict__ cnt) {
    int n = blockIdx.x * blockDim.x + threadIdx.x;
    if (n < NN) atomicAdd(&cnt[batch[n]], 1.0f);
}

__global__ void pool_accum(const float* __restrict__ h,
                           const int* __restrict__ batch,
                           float* __restrict__ out) {
    int t = blockIdx.x * blockDim.x + threadIdx.x;
    if (t >= NN * OUTC) return;
    int n = t >> 8, c = t & (OUTC - 1);
    atomicAdd(&out[batch[n] * OUTC + c], h[t]);
}

__global__ void pool_div(float* __restrict__ out, const float* __restrict__ cnt) {
    int t = blockIdx.x * blockDim.x + threadIdx.x;
    if (t < GG * OUTC) out[t] /= fmaxf(cnt[t >> 8], 1.0f);
}

// ---------------------------------------------------------------------------
extern "C" void kernel_launch(void* const* d_in, const int* in_sizes, int n_in,
                              void* d_out, int out_size, void* d_ws, size_t ws_size,
                              hipStream_t stream) {
    const float* x       = (const float*)d_in[0];
    const int*   ei      = (const int*)  d_in[1];
    const int*   batch   = (const int*)  d_in[2];
    const float* W1      = (const float*)d_in[3];
    const float* asrc1   = (const float*)d_in[4];
    const float* adst1   = (const float*)d_in[5];
    const float* b1      = (const float*)d_in[6];
    const float* ln1g    = (const float*)d_in[7];
    const float* ln1b    = (const float*)d_in[8];
    const float* W2      = (const float*)d_in[9];
    const float* asrc2   = (const float*)d_in[10];
    const float* adst2   = (const float*)d_in[11];
    const float* b2      = (const float*)d_in[12];
    const float* ln2g    = (const float*)d_in[13];
    const float* ln2b    = (const float*)d_in[14];
    float* out = (float*)d_out;

    // ---- workspace layout (bytes, 256-aligned) ----
    size_t off = 0;
    auto alloc = [&](size_t bytes) { size_t o = off; off += (bytes + 255) & ~(size_t)255; return o; };
    size_t o_h1   = alloc((size_t)NN * HC1 * 4);   // GEMM1 output; layer2 reuses it
    size_t o_out1 = alloc((size_t)NN * HC1 * 4);   // layer1 aggregate / LN / GEMM2 input
    size_t o_e    = alloc((size_t)ETOT * HEADS * 4);
    size_t o_es   = alloc((size_t)NN * HEADS * 4);
    size_t o_ed   = alloc((size_t)NN * HEADS * 4);
    size_t o_m    = alloc((size_t)NN * HEADS * 4);
    size_t o_s    = alloc((size_t)NN * HEADS * 4);
    size_t o_cnt  = alloc((size_t)GG * 4);
    (void)ws_size;

    char* ws = (char*)d_ws;
    float* h1   = (float*)(ws + o_h1);
    float* out1 = (float*)(ws + o_out1);
    float* h2   = (float*)(ws + o_h1);                         // alias: h1 dead after layer1 aggr
    float* out2 = (float*)(ws + o_h1 + (size_t)NN * OUTC * 4); // alias: second half of h1 region
    float* ebuf = (float*)(ws + o_e);
    float* es   = (float*)(ws + o_es);
    float* ed   = (float*)(ws + o_ed);
    float* mbuf = (float*)(ws + o_m);
    float* sbuf = (float*)(ws + o_s);
    float* cnt  = (float*)(ws + o_cnt);

    const int TPB = 256;
    const int egrid  = (ETOT + TPB - 1) / TPB;          // 1 thread / edge
    const int eagrid = (ETOT + 7) / 8;                  // 1 wave / edge, 8 edges / block

    // ================= Layer 1 =================
    hipMemsetAsync(out1, 0, (size_t)NN * HC1 * 4, stream);
    init_ms<<<(NN * HEADS + TPB - 1) / TPB, TPB, 0, stream>>>(mbuf, sbuf, NN * HEADS);

    // h1 = x @ W1   [50000,256]x[256,512]
    wmma_gemm_f32<<<dim3(NN / 16, (HC1 / 16) / 4), 128, 0, stream>>>(x, W1, h1, NN, HC1, INC);

    attn_scores<<<(NN * HEADS + TPB - 1) / TPB, TPB, 0, stream>>>(h1, asrc1, adst1, es, ed, HEADS, HID);
    edge_logits<<<egrid, TPB, 0, stream>>>(ei, es, ed, ebuf, mbuf, HEADS);
    edge_exp   <<<egrid, TPB, 0, stream>>>(ei, ebuf, mbuf, sbuf, HEADS);
    edge_aggr  <<<eagrid, TPB, 0, stream>>>(ei, ebuf, sbuf, h1, out1, HEADS, 7 /*log2(128)*/);

    // (agg + b1) -> LN -> ReLU, in place
    ln_kernel<<<(NN + 7) / 8, 256, 0, stream>>>(out1, b1, ln1g, ln1b, HC1, 1);

    // ================= Layer 2 =================
    // h2 = out1 @ W2   [50000,512]x[512,256]   (h2 aliases h1 region)
    wmma_gemm_f32<<<dim3(NN / 16, (OUTC / 16) / 4), 128, 0, stream>>>(out1, W2, h2, NN, OUTC, HC1);

    hipMemsetAsync(out2, 0, (size_t)NN * OUTC * 4, stream);
    init_ms<<<(NN + TPB - 1) / TPB, TPB, 0, stream>>>(mbuf, sbuf, NN);

    attn_scores<<<(NN + TPB - 1) / TPB, TPB, 0, stream>>>(h2, asrc2, adst2, es, ed, 1, OUTC);
    edge_logits<<<egrid, TPB, 0, stream>>>(ei, es, ed, ebuf, mbuf, 1);
    edge_exp   <<<egrid, TPB, 0, stream>>>(ei, ebuf, mbuf, sbuf, 1);
    edge_aggr  <<<eagrid, TPB, 0, stream>>>(ei, ebuf, sbuf, h2, out2, 1, 8 /*log2(256)*/);

    ln_kernel<<<(NN + 7) / 8, 256, 0, stream>>>(out2, b2, ln2g, ln2b, OUTC, 0);

    // ================= Global mean pool =================
    hipMemsetAsync(out, 0, (size_t)GG * OUTC * 4, stream);
    hipMemsetAsync(cnt, 0, (size_t)GG * 4, stream);
    pool_cnt  <<<(NN + TPB - 1) / TPB, TPB, 0, stream>>>(batch, cnt);
    pool_accum<<<((size_t)NN * OUTC + TPB - 1) / TPB, TPB, 0, stream>>>(out2, batch, out);
    pool_div  <<<(GG * OUTC + TPB - 1) / TPB, TPB, 0, stream>>>(out, cnt);
}